// SmoothOperator_16475494547870
// MI455X (gfx1250) — compile-verified
//
#include <hip/hip_runtime.h>
#include <hip/hip_bf16.h>
#include <math.h>

// Separable 4-axis Gaussian filter (sigma=3.5, radius=14, symmetric reflect)
// on a (64, 512, 512, 3) f32 tensor.
//
// MI455X plan (memory-bound: ~1.2 GB / 23.3 TB/s ~= 52 us floor):
//   init : 29-tap kernel + reflect-folded 3x3 channel matrix -> d_ws tail
//   Pass1: conv along W + fused 3x3 channel mix              d_in  -> d_out
//   Pass2: conv along H (cols = flattened w*c, coalesced)    d_out -> d_ws
//   Pass3: conv along B (cols = flattened h*w*c, coalesced)  d_ws  -> d_out
// Each 1D conv = banded-Toeplitz x input-tile matmul on the matrix unit
// via V_WMMA_F32_16X16X4_F32 (fp32 precision, K=44 chunked as 11 x 4).

typedef __attribute__((ext_vector_type(2))) float v2f;
typedef __attribute__((ext_vector_type(8))) float v8f;

#define G_RADIUS 14
#define G_TAPS   29
#define G_SIGMA  3.5
#define NELEM    (64L * 512 * 512 * 3)   // 50,331,648 floats

__device__ __forceinline__ int reflect_idx(int p, int n) {
  // scipy 'reflect' == symmetric; radius(14) < n for every conv axis here,
  // so a single application suffices.
  p = (p < 0) ? (-p - 1) : p;
  p = (p >= n) ? (2 * n - 1 - p) : p;
  return p;
}

// One tiny block computes wtab[0..28] = normalized Gaussian taps and
// wtab[29..37] = 3x3 channel-mix matrix (len-3 axis folded through the
// period-6 symmetric-reflect pattern). Runs once per launch.
__global__ void SmoothOperator_init_weights(float* __restrict__ wtab) {
  const int t = threadIdx.x;
  if (t < G_TAPS) {
    double s = 0.0;
    for (int k = 0; k < G_TAPS; ++k) {
      double x = (double)(k - G_RADIUS) / G_SIGMA;
      s += exp(-0.5 * x * x);
    }
    double x = (double)(t - G_RADIUS) / G_SIGMA;
    wtab[t] = (float)(exp(-0.5 * x * x) / s);
  } else if (t < G_TAPS + 9) {
    const int i = t - G_TAPS;
    const int co = i / 3, ci = i % 3;
    double s = 0.0, acc = 0.0;
    for (int k = 0; k < G_TAPS; ++k) {
      double x = (double)(k - G_RADIUS) / G_SIGMA;
      double w = exp(-0.5 * x * x);
      s += w;
      int p = co + k - G_RADIUS;
      int r = ((p % 6) + 6) % 6;        // 0,1,2,2,1,0 repeating
      r = (r < 3) ? r : 5 - r;
      if (r == ci) acc += w;
    }
    wtab[t] = (float)(acc / s);
  }
}

// Generic 1D Gaussian conv along one axis via WMMA f32 16x16x4.
// One wave computes a 16(axis) x 16(col) output tile.
// chan_mix != 0: B-operand loads apply the 3x3 channel matrix
// (base_off includes the c offset o0; raw pixel base = base_off - o0).
__global__ __launch_bounds__(256)
void SmoothOperator_conv_wmma(const float* __restrict__ in,
                              float* __restrict__ out,
                              const float* __restrict__ wtab,
                              int n_axis, long s_axis, long s_col,
                              int axis_tiles, int col_tiles,
                              int outer0_count, long outer0_stride,
                              int outer1_count, long outer1_stride,
                              int chan_mix) {
  __shared__ float lds_w[G_TAPS];   // 29 Gaussian taps
  __shared__ float lds_m[9];        // 3x3 channel-mix matrix

  if (threadIdx.x < G_TAPS + 9) {
    const float v = wtab[threadIdx.x];
    if (threadIdx.x < G_TAPS) lds_w[threadIdx.x] = v;
    else                      lds_m[threadIdx.x - G_TAPS] = v;
  }
  __syncthreads();

  const int lane = threadIdx.x & 31;
  const int wave = threadIdx.x >> 5;
  long gw = (long)blockIdx.x * (blockDim.x >> 5) + wave;

  int t_axis = (int)(gw % axis_tiles);   gw /= axis_tiles;
  int t_col  = (int)(gw % col_tiles);    gw /= col_tiles;
  int o0     = (int)(gw % outer0_count); gw /= outer0_count;
  int o1     = (int)gw;
  if (o1 >= outer1_count) return;  // wave-uniform; EXEC stays all-1s for WMMA

  const long base_off  = (long)o0 * outer0_stride + (long)o1 * outer1_stride;
  const int  base_axis = t_axis * 16;
  const int  N         = lane & 15;           // column within tile
  const int  col       = t_col * 16 + N;
  const int  M         = lane & 15;           // output row (A operand)
  const int  k0        = (lane < 16) ? 0 : 2; // K split per f32 WMMA layout

  const long col_off = base_off + (long)col * s_col;
  const float mx0 = lds_m[o0 * 3 + 0];
  const float mx1 = lds_m[o0 * 3 + 1];
  const float mx2 = lds_m[o0 * 3 + 2];

  v8f acc = {};
  #pragma unroll
  for (int j = 0; j < 11; ++j) {            // K = 44 = 11 chunks of 4
    const int t0 = 4 * j + k0;
    const int t1 = t0 + 1;

    // A: Toeplitz band of Gaussian taps, A[M, t] = w[t - M]
    const int i0 = t0 - M, i1 = t1 - M;
    v2f a;
    a.x = (i0 >= 0 && i0 < G_TAPS) ? lds_w[i0] : 0.0f;
    a.y = (i1 >= 0 && i1 < G_TAPS) ? lds_w[i1] : 0.0f;

    // B: input gather (reflected axis index), optional fused channel mix
    const int p0 = reflect_idx(base_axis - G_RADIUS + t0, n_axis);
    const int p1 = reflect_idx(base_axis - G_RADIUS + t1, n_axis);
    v2f b;
    if (chan_mix) {
      const float* q0 = in + (col_off - o0) + (long)p0 * s_axis;
      const float* q1 = in + (col_off - o0) + (long)p1 * s_axis;
      b.x = mx0 * q0[0] + mx1 * q0[1] + mx2 * q0[2];
      b.y = mx0 * q1[0] + mx1 * q1[1] + mx2 * q1[2];
    } else {
      b.x = in[col_off + (long)p0 * s_axis];
      b.y = in[col_off + (long)p1 * s_axis];
    }

    // D = A x B + C, fp32 WMMA (16x16x4)
    acc = __builtin_amdgcn_wmma_f32_16x16x4_f32(
        /*neg_a=*/false, a, /*neg_b=*/false, b,
        /*c_mod=*/(short)0, acc, /*reuse_a=*/false, /*reuse_b=*/false);
  }

  // C/D layout: VGPR r holds M = r + (lane<16 ? 0 : 8), N = lane%16
  const int m0 = (lane < 16) ? 0 : 8;
  #pragma unroll
  for (int r = 0; r < 8; ++r) {
    out[base_off + (long)(base_axis + m0 + r) * s_axis + (long)col * s_col] =
        acc[r];
  }
}

extern "C" void kernel_launch(void* const* d_in, const int* in_sizes, int n_in,
                              void* d_out, int out_size, void* d_ws, size_t ws_size,
                              hipStream_t stream) {
  const float* x  = (const float*)d_in[0];
  float* out      = (float*)d_out;
  float* ws       = (float*)d_ws;       // needs >= 201 MB + 152 B
  float* wtab     = ws + NELEM;         // 38-float weight table in ws tail

  const int  B = 64, H = 512, W = 512;
  const long SB = (long)H * W * 3;      // 786432
  const long SH = (long)W * 3;          // 1536
  const long SW = 3;

  dim3 blk(256);   // 8 waves/block, one 16x16 tile per wave

  // Init: compute taps + channel matrix once (1 block)
  SmoothOperator_init_weights<<<1, 64, 0, stream>>>(wtab);

  // Pass 1: conv along W + fused channel mix:  d_in -> d_out
  // axis=W(512, stride 3), col=H(512, stride 1536), outer: c(3,1), b(64,SB)
  {
    long waves = 32L * 32 * 3 * 64;               // 196608
    SmoothOperator_conv_wmma<<<(int)(waves / 8), blk, 0, stream>>>(
        x, out, wtab, W, SW, SH, 32, 32, 3, 1L, 64, SB, /*chan_mix=*/1);
  }
  // Pass 2: conv along H:  d_out -> d_ws
  // axis=H(512, stride 1536), col=(w,c) flat (1536, stride 1), outer: b(64,SB)
  {
    long waves = 32L * 96 * 64;                   // 196608
    SmoothOperator_conv_wmma<<<(int)(waves / 8), blk, 0, stream>>>(
        out, ws, wtab, H, SH, 1L, 32, 96, 64, SB, 1, 0L, /*chan_mix=*/0);
  }
  // Pass 3: conv along batch:  d_ws -> d_out
  // axis=B(64, stride SB), col=(h,w,c) flat (786432, stride 1)
  {
    long waves = 4L * 49152;                      // 196608
    SmoothOperator_conv_wmma<<<(int)(waves / 8), blk, 0, stream>>>(
        ws, out, wtab, B, SB, 1L, 4, 49152, 1, 0L, 1, 0L, /*chan_mix=*/0);
  }
}